// TreeSpotLocalizationLoss_37503654429214
// MI455X (gfx1250) — compile-verified
//
#include <hip/hip_runtime.h>
#include <hip/hip_bf16.h>
#include <math.h>

typedef __attribute__((ext_vector_type(2))) float v2f;
typedef __attribute__((ext_vector_type(8))) float v8f;

#define NBLK 1024   // partial-reduction blocks (multiple of 64 for WMMA tiles)
#define NTHR 256    // 8 wave32s per block
#define NACC 5      // t_sum, p_sum, TP, x2_sum, focal_sum

// ---------------------------------------------------------------------------
// Per-element math, minimized transcendentals:
//   e  = exp(-x)                    (v_exp_f32)
//   s  = sigmoid(x) = 1/(1+e)       (v_rcp_f32)
//   L  = log(s) = log p             (v_log_f32)
//   -log p      = -L      (bce term for t==1, clamped at 100)
//   -log(1-p)   = x - L   (bce term for t==0, clamped at 100)
//   pt = t ? s : (1-s)   (binary mask -> no extra exp needed)
// ---------------------------------------------------------------------------
__device__ __forceinline__ void accum_elem(float x, float t,
                                           float& s_t, float& s_p, float& s_tp,
                                           float& s_x2, float& s_f) {
  float e = __expf(-x);
  float s = __builtin_amdgcn_rcpf(1.0f + e);       // sigmoid(x)
  float L = __logf(s);                             // log p
  float a = fminf(-L, 100.0f);                     // clamped -log p
  float b = fminf(x - L, 100.0f);                  // clamped -log(1-p)
  float bce = b + t * (a - b);
  float pt  = (1.0f - s) + t * (2.0f * s - 1.0f);  // t ? s : 1-s
  float om  = 1.0f - pt;
  s_f  += om * om * bce;                           // focal term (gamma=2, alpha=1)
  s_t  += t;
  s_p  += s;
  s_tp += s * t;
  s_x2 += x * x;
}

__global__ void loss_partial_kernel(const float* __restrict__ pred,
                                    const float* __restrict__ targ,
                                    float* __restrict__ part, int n) {
  float s_t = 0.0f, s_p = 0.0f, s_tp = 0.0f, s_x2 = 0.0f, s_f = 0.0f;

  const int tid    = blockIdx.x * blockDim.x + threadIdx.x;
  const int stride = gridDim.x * blockDim.x;
  const int n4     = n >> 2;  // n is a multiple of 4 (8*1024*1024)

  const float4* __restrict__ p4 = reinterpret_cast<const float4*>(pred);
  const float4* __restrict__ t4 = reinterpret_cast<const float4*>(targ);

  for (int i = tid; i < n4; i += stride) {
    // hint the next grid-stride line into cache (global_prefetch_b8)
    __builtin_prefetch(&p4[i + stride], 0, 0);
    __builtin_prefetch(&t4[i + stride], 0, 0);
    float4 xv = p4[i];
    float4 tv = t4[i];
    accum_elem(xv.x, tv.x, s_t, s_p, s_tp, s_x2, s_f);
    accum_elem(xv.y, tv.y, s_t, s_p, s_tp, s_x2, s_f);
    accum_elem(xv.z, tv.z, s_t, s_p, s_tp, s_x2, s_f);
    accum_elem(xv.w, tv.w, s_t, s_p, s_tp, s_x2, s_f);
  }

  // wave32 butterfly reduction
  #pragma unroll
  for (int off = 16; off > 0; off >>= 1) {
    s_t  += __shfl_xor(s_t,  off, 32);
    s_p  += __shfl_xor(s_p,  off, 32);
    s_tp += __shfl_xor(s_tp, off, 32);
    s_x2 += __shfl_xor(s_x2, off, 32);
    s_f  += __shfl_xor(s_f,  off, 32);
  }

  __shared__ float red[NTHR / 32][NACC];
  const int lane = threadIdx.x & 31;
  const int wave = threadIdx.x >> 5;
  if (lane == 0) {
    red[wave][0] = s_t;  red[wave][1] = s_p;  red[wave][2] = s_tp;
    red[wave][3] = s_x2; red[wave][4] = s_f;
  }
  __syncthreads();

  if (threadIdx.x == 0) {
    float r0 = 0.0f, r1 = 0.0f, r2 = 0.0f, r3 = 0.0f, r4 = 0.0f;
    #pragma unroll
    for (int w = 0; w < NTHR / 32; ++w) {
      r0 += red[w][0]; r1 += red[w][1]; r2 += red[w][2];
      r3 += red[w][3]; r4 += red[w][4];
    }
    // SoA layout: part[acc * gridDim.x + block]
    part[0 * gridDim.x + blockIdx.x] = r0;
    part[1 * gridDim.x + blockIdx.x] = r1;
    part[2 * gridDim.x + blockIdx.x] = r2;
    part[3 * gridDim.x + blockIdx.x] = r3;
    part[4 * gridDim.x + blockIdx.x] = r4;
  }
}

// ---------------------------------------------------------------------------
// One-wave combine: WMMA (16x16x4 f32) as a reduction engine.
// A 64-float tile is laid out per the ISA A-matrix layout:
//   A[m][k] = tile[m*4 + k], lane L holds m = L%16, k = (L/16)*2 + {0,1}.
// D = A x ones + D  ->  every D column holds the 16 row sums; tile total =
// (sum of 8 D regs on any lane<16) + (same on its lane^16 partner).
// ---------------------------------------------------------------------------
__global__ void loss_combine_kernel(const float* __restrict__ part,
                                    float* __restrict__ out,
                                    int nb, float inv_n) {
  const int lane = threadIdx.x;                       // 0..31, full wave
  const int aoff = (lane & 15) * 4 + (lane >> 4) * 2; // A-layout offset in tile

  v2f ones; ones[0] = 1.0f; ones[1] = 1.0f;

  float sums[NACC];
  #pragma unroll
  for (int j = 0; j < NACC; ++j) {
    const float* base = part + j * nb;
    v8f d = {};
    for (int tile = 0; tile < nb; tile += 64) {
      v2f a;
      a[0] = base[tile + aoff + 0];
      a[1] = base[tile + aoff + 1];
      d = __builtin_amdgcn_wmma_f32_16x16x4_f32(
          /*neg_a=*/false, a, /*neg_b=*/false, ones,
          /*c_mod=*/(short)0, d, /*reuse_a=*/false, /*reuse_b=*/false);
    }
    float h = d[0] + d[1] + d[2] + d[3] + d[4] + d[5] + d[6] + d[7];
    sums[j] = h + __shfl_xor(h, 16, 32);  // rows 0-7 half + rows 8-15 half
  }

  if (lane == 0) {
    const float EPS = 1e-6f;
    float t_sum = sums[0], p_sum = sums[1], TP = sums[2];
    float x2    = sums[3], fsum  = sums[4];

    float FPv = p_sum - TP;
    float FNv = t_sum - TP;

    float dice      = (2.0f * TP + EPS) / (p_sum + t_sum + EPS);
    float dice_loss = (t_sum == 0.0f) ? (x2 * inv_n) : (1.0f - dice);

    float tv           = (TP + EPS) / (TP + 0.5f * FPv + 0.5f * FNv + EPS);
    float tversky_loss = 1.0f - tv;

    float focal = fsum * inv_n;

    float ftv           = (TP + EPS) / (TP + 0.7f * FPv + 0.3f * FNv + EPS);
    float focal_tversky = powf(fmaxf(1.0f - ftv, 0.0f), 0.75f);  // (1-ftv)^(1/(4/3))

    out[0] = 0.6f * dice_loss + 0.2f * tversky_loss +
             0.2f * focal + 0.2f * focal_tversky;
  }
}

extern "C" void kernel_launch(void* const* d_in, const int* in_sizes, int n_in,
                              void* d_out, int out_size, void* d_ws, size_t ws_size,
                              hipStream_t stream) {
  const float* pred = (const float*)d_in[0];
  const float* targ = (const float*)d_in[1];
  float* out  = (float*)d_out;
  float* part = (float*)d_ws;     // NACC * NBLK floats = 20 KB
  const int n = in_sizes[0];

  loss_partial_kernel<<<NBLK, NTHR, 0, stream>>>(pred, targ, part, n);
  loss_combine_kernel<<<1, 32, 0, stream>>>(part, out, NBLK, 1.0f / (float)n);
}